// SoniaLayer_53429393162997
// MI455X (gfx1250) — compile-verified
//
#include <hip/hip_runtime.h>
#include <hip/hip_bf16.h>

typedef __attribute__((ext_vector_type(2))) float v2f;
typedef __attribute__((ext_vector_type(4))) float v4f;
typedef __attribute__((ext_vector_type(8))) float v8f;

#define NROWS (1u << 20)   // OUTPUT_FEATURES
#define KDIM  256          // INPUT_FEATURES
#define WAVES_PER_BLOCK 8  // 256 threads, wave32

// One wave computes a 16-row tile. Per 8-k block: one global_load_b128 of W
// (lane half h covers k0+4h..k0+4h+3) feeds TWO V_WMMA_F32_16X16X4_F32 ops
// (first halves of the quads -> slots {k0,k0+1 | k0+4,k0+5}; second halves ->
// {k0+2,k0+3 | k0+6,k0+7}), with x supplied from LDS via a matching
// ds_load_b128. sum(w^2)/sum(x^2) ride along as v2f pk-FMA accumulators on the
// same naturally-paired registers (no operand-assembly movs).
__global__ __launch_bounds__(256) void sonia_l2_tanh_kernel(
    const float* __restrict__ x,      // [1, 256]
    const float* __restrict__ w,      // [1M, 256] row-major
    float* __restrict__ out)          // [1M]
{
    __shared__ float xs[KDIM];
    const int tid = threadIdx.x;
    xs[tid] = x[tid];                 // stage the input row once per block
    __syncthreads();

    const int lane   = tid & 31;
    const int waveId = tid >> 5;
    const int half   = lane >> 4;     // K-half selector for the 16x4 A layout
    const int m      = lane & 15;     // row-in-tile for A loads

    const int tile = (int)blockIdx.x * WAVES_PER_BLOCK + waveId;
    const int row0 = tile * 16;

    // Lane's A pointer: row (row0+m), starting at K offset 4*half.
    const float* wrow = w + (size_t)(row0 + m) * KDIM + 4 * half;
    const float* xrow = xs + 4 * half;

    v8f acc = {};          // WMMA accumulator: dot(row, x) replicated across N
    v2f sww2 = {0.f, 0.f}; // per-lane packed partial sum of w^2
    v2f sxx2 = {0.f, 0.f}; // per-lane packed partial sum of x^2

#pragma unroll 8
    for (int k0 = 0; k0 < KDIM; k0 += 8) {
        v4f a = *(const v4f*)(wrow + k0);   // global b128: W[row][k0+4h..+3]
        v4f b = *(const v4f*)(xrow + k0);   // LDS b128 broadcast: x[k0+4h..+3]
        v2f a01 = {a.x, a.y}, a23 = {a.z, a.w};
        v2f b01 = {b.x, b.y}, b23 = {b.z, b.w};
        acc = __builtin_amdgcn_wmma_f32_16x16x4_f32(
            false, a01, false, b01, (short)0, acc, false, false);
        acc = __builtin_amdgcn_wmma_f32_16x16x4_f32(
            false, a23, false, b23, (short)0, acc, false, false);
        sww2 += a01 * a01;
        sww2 += a23 * a23;
        sxx2 += b01 * b01;
        sxx2 += b23 * b23;
    }

    // Horizontal + cross-half reduce: lane L then holds full sum(w^2) of row
    // (L&15); sxx becomes the full sum(x^2), uniform across lanes.
    float sww = sww2.x + sww2.y;
    float sxx = sxx2.x + sxx2.y;
    sww += __shfl_xor(sww, 16);
    sxx += __shfl_xor(sxx, 16);

    // D layout: VGPR j holds M=j (lanes 0-15) / M=j+8 (lanes 16-31), identical
    // across N. Emit 16 outputs; two active lanes per store (store traffic is
    // negligible vs the 1 GB weight stream).
    float* obase = out + row0;
#pragma unroll
    for (int j = 0; j < 8; ++j) {
        const int r = half ? (j + 8) : j;       // row this half's D holds
        const float ss = __shfl(sww, r);        // row r's sum(w^2) from lane r
        float d2 = ss - 2.0f * acc[j] + sxx;
        d2 = d2 < 0.0f ? 0.0f : d2;             // guard fp cancellation
        const float v = tanhf(sqrtf(d2));
        if (m == 0) obase[r] = v;               // lanes 0 and 16 write
    }
}

extern "C" void kernel_launch(void* const* d_in, const int* in_sizes, int n_in,
                              void* d_out, int out_size, void* d_ws, size_t ws_size,
                              hipStream_t stream) {
    const float* x = (const float*)d_in[0];   // input [1,256]
    const float* w = (const float*)d_in[1];   // weight [1M,256]
    float* out = (float*)d_out;               // [1M]
    (void)in_sizes; (void)n_in; (void)out_size; (void)d_ws; (void)ws_size;

    const int tiles = NROWS / 16;                       // 65536 waves
    const int blocks = tiles / WAVES_PER_BLOCK;         // 8192 blocks
    sonia_l2_tanh_kernel<<<blocks, 32 * WAVES_PER_BLOCK, 0, stream>>>(x, w, out);
}